// HRNEncoder_45320494907744
// MI455X (gfx1250) — compile-verified
//
#include <hip/hip_runtime.h>

// ---------------------------------------------------------------------------
// HRN GNN encoder for MI455X (gfx1250, wave32).
// Bandwidth-bound workload (~1 GB traffic, ~6 GFLOP) -> keep fp32 end-to-end
// and use V_WMMA_F32_16X16X4_F32 for every GEMM.
// ---------------------------------------------------------------------------

typedef float v2f __attribute__((ext_vector_type(2)));
typedef float v8f __attribute__((ext_vector_type(8)));

#define H_DIM 64      // hidden width (= GEMM N dimension, fixed)
#define F_DIM 128     // input feature width
#define B_GRAPHS 1024 // num_graphs (fixed by reference setup)
#define BN_EPS 1e-5f

// ---------------- zero workspace (accumulation buffers) ----------------
__global__ void k_zero(float* __restrict__ p, long n) {
  long t = (long)blockIdx.x * blockDim.x + threadIdx.x;
  long stride = (long)gridDim.x * blockDim.x;
  for (; t < n; t += stride) p[t] = 0.0f;
}

// ---------------- scatter-add: out[dst[i]] += X[src[i]], float4 lanes ------
// c4shift: log2(cols/4); cols = 4<<c4shift (32 for F=128, 16 for H=64)
__global__ void k_scatter_add(const float* __restrict__ X,
                              const int* __restrict__ src,
                              const int* __restrict__ dst,
                              float* __restrict__ out,
                              long nEdges, int c4shift) {
  long t = (long)blockIdx.x * blockDim.x + threadIdx.x;
  if (t >= (nEdges << c4shift)) return;
  long i = t >> c4shift;
  int c4 = 1 << c4shift;
  int c = (int)(t & (c4 - 1));
  // streaming prefetch of the source feature stream (global_prefetch_b8)
  __builtin_prefetch((const char*)X + (t << 4) + 8192, 0, 1);
  float4 v = ((const float4*)X)[(size_t)src[i] * c4 + c];
  float* o = out + ((size_t)dst[i] * c4 + c) * 4;
  atomicAdd(o + 0, v.x);
  atomicAdd(o + 1, v.y);
  atomicAdd(o + 2, v.z);
  atomicAdd(o + 3, v.w);
}

// ---------------- pooled[batch[i]][colOff + :64] += X[i] -------------------
__global__ void k_pool_add(const float* __restrict__ X,
                           const int* __restrict__ batch,
                           float* __restrict__ pooled,
                           long R, int colOff) {
  long t = (long)blockIdx.x * blockDim.x + threadIdx.x;
  if (t >= (R << 4)) return;
  long i = t >> 4;
  int c = (int)(t & 15);
  float4 v = ((const float4*)X)[(size_t)i * 16 + c];
  float* o = pooled + (size_t)batch[i] * (2 * H_DIM) + colOff + c * 4;
  atomicAdd(o + 0, v.x);
  atomicAdd(o + 1, v.y);
  atomicAdd(o + 2, v.z);
  atomicAdd(o + 3, v.w);
}

// ---------------- C[M,64] = A[M,K] @ W[K,64] + bias  (fp32 WMMA) -----------
// One wave computes a 16x64 tile: 4 accumulators, K stepped by 4.
// A 16x4 layout: lanes 0-15 M=0..15, v0 holds K=k0 (lo half)/k0+2 (hi half),
// v1 holds K=k0+1 / k0+3.  C/D: v[i] = row (half*8+i), col = tile*16 + (lane&15).
__device__ __forceinline__ void store_tile(float* __restrict__ C, int r0, int n,
                                           int half, const v8f& acc, float bv) {
#pragma unroll
  for (int i = 0; i < 8; ++i)
    C[(size_t)(r0 + half * 8 + i) * H_DIM + n] = acc[i] + bv;
}

__global__ void k_gemm_bias(const float* __restrict__ A,
                            const float* __restrict__ W,
                            const float* __restrict__ bias,
                            float* __restrict__ C,
                            int M, int K) {
  int wave = (int)(((long)blockIdx.x * blockDim.x + threadIdx.x) >> 5);
  int lane = threadIdx.x & 31;
  int r0 = wave << 4;
  if (r0 >= M) return;          // whole wave exits together: EXEC stays all-1s
  int half = lane >> 4;         // 0: lanes 0-15, 1: lanes 16-31
  int l = lane & 15;

  v8f acc0 = {}, acc1 = {}, acc2 = {}, acc3 = {};
  const float* Ar = A + (size_t)(r0 + l) * K;

  for (int k0 = 0; k0 < K; k0 += 4) {
    v2f a;
    a.x = Ar[k0 + 2 * half];
    a.y = Ar[k0 + 2 * half + 1];
    const float* Wk = W + (size_t)(k0 + 2 * half) * H_DIM + l;
    v2f b0, b1, b2, b3;
    b0.x = Wk[0];  b0.y = Wk[H_DIM];
    b1.x = Wk[16]; b1.y = Wk[16 + H_DIM];
    b2.x = Wk[32]; b2.y = Wk[32 + H_DIM];
    b3.x = Wk[48]; b3.y = Wk[48 + H_DIM];
    acc0 = __builtin_amdgcn_wmma_f32_16x16x4_f32(false, a, false, b0, (short)0, acc0, false, false);
    acc1 = __builtin_amdgcn_wmma_f32_16x16x4_f32(false, a, false, b1, (short)0, acc1, false, false);
    acc2 = __builtin_amdgcn_wmma_f32_16x16x4_f32(false, a, false, b2, (short)0, acc2, false, false);
    acc3 = __builtin_amdgcn_wmma_f32_16x16x4_f32(false, a, false, b3, (short)0, acc3, false, false);
  }

  store_tile(C, r0, 0 + l,  half, acc0, bias[0 + l]);
  store_tile(C, r0, 16 + l, half, acc1, bias[16 + l]);
  store_tile(C, r0, 32 + l, half, acc2, bias[32 + l]);
  store_tile(C, r0, 48 + l, half, acc3, bias[48 + l]);
}

// ---------------- per-column sum / sum-of-squares over R rows (64 cols) ----
__global__ void k_col_stats(const float* __restrict__ Hm, long R,
                            float* __restrict__ sum, float* __restrict__ sumsq) {
  int col = threadIdx.x & 63;
  long rowStart = (long)blockIdx.x * (blockDim.x >> 6) + (threadIdx.x >> 6);
  long rowStride = (long)gridDim.x * (blockDim.x >> 6);
  float s = 0.f, q = 0.f;
  for (long r = rowStart; r < R; r += rowStride) {
    float v = Hm[(size_t)r * H_DIM + col];
    s += v;
    q += v * v;
  }
  atomicAdd(&sum[col], s);
  atomicAdd(&sumsq[col], q);
}

// ---------------- BatchNorm (training stats) + ReLU, in place --------------
__global__ void k_bn_relu(float* __restrict__ Hm, long R,
                          const float* __restrict__ sum,
                          const float* __restrict__ sumsq,
                          const float* __restrict__ g,
                          const float* __restrict__ be) {
  long t = (long)blockIdx.x * blockDim.x + threadIdx.x;
  long total = R * H_DIM;
  if (t >= total) return;
  int col = (int)(t & (H_DIM - 1));
  float invR = 1.0f / (float)R;
  float m = sum[col] * invR;
  float var = sumsq[col] * invR - m * m;
  var = var > 0.f ? var : 0.f;
  float sc = g[col] * rsqrtf(var + BN_EPS);
  float sh = be[col] - m * sc;
  float h = Hm[t] * sc + sh;
  Hm[t] = h > 0.f ? h : 0.f;
}

// ---------------------------------------------------------------------------
extern "C" void kernel_launch(void* const* d_in, const int* in_sizes, int n_in,
                              void* d_out, int out_size, void* d_ws, size_t ws_size,
                              hipStream_t stream) {
  const float* x      = (const float*)d_in[0];
  const int*   e1_dst = (const int*)d_in[1];
  const int*   e1_src = (const int*)d_in[2];
  const int*   e2_dst = (const int*)d_in[3];
  const int*   e2_src = (const int*)d_in[4];
  const int*   batch1 = (const int*)d_in[5];
  const int*   batch2 = (const int*)d_in[6];
  const float* W1a = (const float*)d_in[7];
  const float* b1a = (const float*)d_in[8];
  const float* g1a = (const float*)d_in[9];
  const float* be1a = (const float*)d_in[10];
  const float* W1b = (const float*)d_in[11];
  const float* b1b = (const float*)d_in[12];
  const float* g1b = (const float*)d_in[13];
  const float* be1b = (const float*)d_in[14];
  const float* W2a = (const float*)d_in[15];
  const float* b2a = (const float*)d_in[16];
  const float* g2a = (const float*)d_in[17];
  const float* be2a = (const float*)d_in[18];
  const float* W2b = (const float*)d_in[19];
  const float* b2b = (const float*)d_in[20];
  const float* g2b = (const float*)d_in[21];
  const float* be2b = (const float*)d_in[22];
  const float* Wout = (const float*)d_in[23];
  const float* bout = (const float*)d_in[24];

  const long N0 = in_sizes[1];   // 1,000,000 edges/nodes layer 0
  const long N1 = in_sizes[3];   // 200,000
  const long N2 = in_sizes[6];   // 40,000

  // -------- workspace layout (floats). Accumulation buffers first so one
  // zeroing pass covers everything that must start at 0 on every call. ------
  float* ws = (float*)d_ws;
  float* agg1   = ws;                                      // [N1,128]
  float* agg2   = agg1 + (size_t)N1 * F_DIM;               // [N2,64]
  float* pooled = agg2 + (size_t)N2 * H_DIM;               // [B,128]
  float* stats  = pooled + (size_t)B_GRAPHS * (2 * H_DIM); // 8 x 64
  float* h1a    = stats + 512;                             // [N1,64]
  float* x1     = h1a + (size_t)N1 * H_DIM;                // [N1,64]
  float* h2a    = x1 + (size_t)N1 * H_DIM;                 // [N2,64]
  float* x2     = h2a + (size_t)N2 * H_DIM;                // [N2,64]
  long zeroN = (long)(stats + 512 - ws);

  auto nb = [](long total, int bs) { return (int)((total + bs - 1) / bs); };
  const int BS = 256; // 8 waves / workgroup

  // 1) zero accumulation buffers
  k_zero<<<2048, BS, 0, stream>>>(ws, zeroN);

  // 2) conv1 aggregation: agg1[e1_dst] += x[e1_src]   (dominant: 512MB read)
  k_scatter_add<<<nb(N0 << 5, BS), BS, 0, stream>>>(x, e1_src, e1_dst, agg1, N0, 5);

  // 3) MLP1 layer a: h1a = agg1 @ W1a + b1a ; BN ; ReLU
  k_gemm_bias<<<nb(N1 / 16, 8), BS, 0, stream>>>(agg1, W1a, b1a, h1a, (int)N1, F_DIM);
  k_col_stats<<<1024, BS, 0, stream>>>(h1a, N1, stats + 0, stats + 64);
  k_bn_relu<<<nb(N1 * H_DIM, BS), BS, 0, stream>>>(h1a, N1, stats + 0, stats + 64, g1a, be1a);

  // 4) MLP1 layer b: x1 = relu(bn(h1a @ W1b + b1b))
  k_gemm_bias<<<nb(N1 / 16, 8), BS, 0, stream>>>(h1a, W1b, b1b, x1, (int)N1, H_DIM);
  k_col_stats<<<1024, BS, 0, stream>>>(x1, N1, stats + 128, stats + 192);
  k_bn_relu<<<nb(N1 * H_DIM, BS), BS, 0, stream>>>(x1, N1, stats + 128, stats + 192, g1b, be1b);

  // 5) conv2 aggregation: agg2[e2_dst] += x1[e2_src]
  k_scatter_add<<<nb(N1 << 4, BS), BS, 0, stream>>>(x1, e2_src, e2_dst, agg2, N1, 4);

  // 6) MLP2 layer a
  k_gemm_bias<<<nb(N2 / 16, 8), BS, 0, stream>>>(agg2, W2a, b2a, h2a, (int)N2, H_DIM);
  k_col_stats<<<1024, BS, 0, stream>>>(h2a, N2, stats + 256, stats + 320);
  k_bn_relu<<<nb(N2 * H_DIM, BS), BS, 0, stream>>>(h2a, N2, stats + 256, stats + 320, g2a, be2a);

  // 7) MLP2 layer b -> x2
  k_gemm_bias<<<nb(N2 / 16, 8), BS, 0, stream>>>(h2a, W2b, b2b, x2, (int)N2, H_DIM);
  k_col_stats<<<1024, BS, 0, stream>>>(x2, N2, stats + 384, stats + 448);
  k_bn_relu<<<nb(N2 * H_DIM, BS), BS, 0, stream>>>(x2, N2, stats + 384, stats + 448, g2b, be2b);

  // 8) global sum pooling straight into concatenated [B, 128] buffer
  k_pool_add<<<nb(N1 << 4, BS), BS, 0, stream>>>(x1, batch1, pooled, N1, 0);
  k_pool_add<<<nb(N2 << 4, BS), BS, 0, stream>>>(x2, batch2, pooled, N2, H_DIM);

  // 9) dim_align: out[B,64] = pooled[B,128] @ Wout + bout
  k_gemm_bias<<<nb(B_GRAPHS / 16, 8), BS, 0, stream>>>(pooled, Wout, bout, (float*)d_out,
                                                       B_GRAPHS, 2 * H_DIM);
}